// CombinedLoss_13838384628301
// MI455X (gfx1250) — compile-verified
//
#include <hip/hip_runtime.h>

typedef __attribute__((ext_vector_type(2))) float v2f;
typedef __attribute__((ext_vector_type(8))) float v8f;

#define EPSL 1e-5f

// Per-element contribution: numerically stable BCE-with-logits + sigmoid terms.
// e = exp(-|x|):  softplus(x) = max(x,0) + log(1+e)
//                 sigmoid(x)  = (x>=0) ? 1/(1+e) : e/(1+e)
__device__ __forceinline__ void accum_elem(float x, float t,
                                           float& ce, float& ps, float& pt, float& ts) {
    float ax  = __builtin_fabsf(x);
    float e   = __expf(-ax);
    float opp = 1.0f + e;
    float r   = __builtin_amdgcn_rcpf(opp);     // v_rcp_f32
    float p   = (x >= 0.0f) ? r : e * r;        // sigmoid(x)
    float sp  = fmaxf(x, 0.0f) + __logf(opp);   // softplus(x)
    ce += sp - x * t;
    ps += p;
    pt += p * t;
    ts += t;
}

// Streaming partial-reduction kernel.
// Writes part[j * gridDim.x + blockIdx.x] for j in {ce, sum_p, sum_p*t, sum_t}.
__global__ __launch_bounds__(256) void loss_partial(const float* __restrict__ pred,
                                                    const float* __restrict__ tgt,
                                                    float* __restrict__ part,
                                                    long long nvec, long long n) {
    const float4* __restrict__ p4 = (const float4*)pred;
    const float4* __restrict__ t4 = (const float4*)tgt;
    long long gid    = (long long)blockIdx.x * blockDim.x + threadIdx.x;
    long long stride = (long long)gridDim.x * blockDim.x;

    float ce = 0.0f, ps = 0.0f, pt = 0.0f, ts = 0.0f;

    for (long long i = gid; i < nvec; i += stride) {
        // Look-ahead prefetch (gfx1250 global_prefetch_b8); speculative, OOB-safe.
        __builtin_prefetch(p4 + i + stride, 0, 0);
        __builtin_prefetch(t4 + i + stride, 0, 0);
        float4 x = p4[i];          // global_load_b128
        float4 t = t4[i];          // global_load_b128
        accum_elem(x.x, t.x, ce, ps, pt, ts);
        accum_elem(x.y, t.y, ce, ps, pt, ts);
        accum_elem(x.z, t.z, ce, ps, pt, ts);
        accum_elem(x.w, t.w, ce, ps, pt, ts);
    }
    // Scalar tail (n not multiple of 4) — handled by one thread, reconverges after.
    if (gid == 0) {
        for (long long i = nvec * 4; i < n; ++i)
            accum_elem(pred[i], tgt[i], ce, ps, pt, ts);
    }

    // ---- Wave32 reduction of the 4 accumulators via V_WMMA_F32_16X16X4_F32 ----
    // A (16x4 f32, 2 VGPRs/lane): lane m (0-15): {A[m][0],A[m][1]}; lane m+16: {A[m][2],A[m][3]}.
    // B is a column selector: col j collects accumulator j across both lane halves.
    //   WMMA#1: A=(ce,ps), B[0][0]=B[2][0]=1 (col0<-ce), B[1][1]=B[3][1]=1 (col1<-ps)
    //   WMMA#2: A=(pt,ts), cols 2,3; C chained from WMMA#1.
    // Result: D[m][j] = acc_j[m] + acc_j[m+16]. Sum of 8 D-VGPRs per lane gives the
    // half-column sums; shfl_xor(16) completes: lane j (j<4) holds total of acc_j.
    const int lane = threadIdx.x & 31;
    const int wave = threadIdx.x >> 5;
    const int lm   = lane & 15;

    v2f a01; a01.x = ce; a01.y = ps;
    v2f a23; a23.x = pt; a23.y = ts;
    v2f b01; b01.x = (lm == 0) ? 1.0f : 0.0f; b01.y = (lm == 1) ? 1.0f : 0.0f;
    v2f b23; b23.x = (lm == 2) ? 1.0f : 0.0f; b23.y = (lm == 3) ? 1.0f : 0.0f;

    v8f c = {};
    c = __builtin_amdgcn_wmma_f32_16x16x4_f32(false, a01, false, b01, (short)0, c, false, false);
    c = __builtin_amdgcn_wmma_f32_16x16x4_f32(false, a23, false, b23, (short)0, c, false, false);

    float s = c[0] + c[1] + c[2] + c[3] + c[4] + c[5] + c[6] + c[7];
    s += __shfl_xor(s, 16, 32);   // lane j + lane j+16 -> full 32-lane total per column

    __shared__ float lds[8 * 4];  // 8 waves x 4 sums
    if (lane < 4) lds[wave * 4 + lane] = s;
    __syncthreads();

    if (threadIdx.x < 4) {
        float tot = 0.0f;
        #pragma unroll
        for (int w = 0; w < 8; ++w) tot += lds[w * 4 + threadIdx.x];
        part[(long long)threadIdx.x * gridDim.x + blockIdx.x] = tot;
    }
}

// Final reduction of 4 x nb partials + scalar combine. One block, 4 waves,
// wave j reduces sum j. Deterministic (fixed order within each lane, fixed tree).
__global__ __launch_bounds__(128) void loss_final(const float* __restrict__ part,
                                                  float* __restrict__ out,
                                                  int nb, float invN) {
    const int wave = threadIdx.x >> 5;
    const int lane = threadIdx.x & 31;
    const float* src = part + (long long)wave * nb;

    float acc = 0.0f;
    for (int i = lane; i < nb; i += 32) acc += src[i];
    #pragma unroll
    for (int off = 16; off > 0; off >>= 1) acc += __shfl_xor(acc, off, 32);

    __shared__ float sums[4];
    if (lane == 0) sums[wave] = acc;
    __syncthreads();

    if (threadIdx.x == 0) {
        float ce_sum = sums[0], p_sum = sums[1], pt_sum = sums[2], t_sum = sums[3];
        float ce   = ce_sum * invN;                                    // mean BCE
        float dice = 1.0f - (2.0f * pt_sum + EPSL) / (p_sum + t_sum + EPSL);
        out[0] = 0.5f * dice + 0.5f * ce;
    }
}

extern "C" void kernel_launch(void* const* d_in, const int* in_sizes, int n_in,
                              void* d_out, int out_size, void* d_ws, size_t ws_size,
                              hipStream_t stream) {
    const float* pred = (const float*)d_in[0];
    const float* tgt  = (const float*)d_in[1];
    // d_in[2] = n_classes (==2, binary path) — formula is fixed for that path.

    long long n    = (long long)in_sizes[0];
    long long nvec = n >> 2;

    int NB = 4096;                                   // 4096 blocks x 256 thr: 1M threads
    size_t need = (size_t)4 * NB * sizeof(float);    // 64 KiB of partials
    if (ws_size < need) {
        NB = (int)(ws_size / (4 * sizeof(float)));
        if (NB < 1) NB = 1;
    }
    float* part = (float*)d_ws;

    loss_partial<<<NB, 256, 0, stream>>>(pred, tgt, part, nvec, n);
    loss_final<<<1, 128, 0, stream>>>(part, (float*)d_out, NB, 1.0f / (float)n);
}